// GATNet_80831284511065
// MI455X (gfx1250) — compile-verified
//
#include <hip/hip_runtime.h>

#define NN   20000
#define EE   120000
#define BB   100
#define NPGC 200
#define HHC  8
#define DDC  128
#define HD   1024

typedef float v2f __attribute__((ext_vector_type(2)));
typedef float v8f __attribute__((ext_vector_type(8)));

__device__ __forceinline__ float lrelu_f(float x) { return x >= 0.0f ? x : 0.01f * x; }

// monotonic float<->uint encoding for atomic max over floats (incl. negatives)
__device__ __forceinline__ unsigned ford(float x) {
  unsigned u = __float_as_uint(x);
  return (u & 0x80000000u) ? ~u : (u | 0x80000000u);
}
__device__ __forceinline__ float fdec(unsigned u) {
  unsigned v = (u & 0x80000000u) ? (u & 0x7FFFFFFFu) : ~u;
  return __uint_as_float(v);
}

// ---------------------------------------------------------------------------
// ht[N,1024] = h[N,KDIM] @ Wn[KDIM,1024]   (fp32 WMMA 16x16x4)
// grid = (N/16, 1024/256), block = 256 (8 waves).
// Each wave computes a 16x32 output block (two 16x16 WMMA tiles) so each
// A fragment feeds two WMMAs. KDIM is compile-time: guards fold away for
// KDIM=128 -> fully unrolled straight-line loop of ds_load_b64 + wmma.
// ---------------------------------------------------------------------------
template <int KDIM>
__global__ __launch_bounds__(256)
void gat_node_gemm(const float* __restrict__ h, const float* __restrict__ Wn,
                   float* __restrict__ ht)
{
  constexpr int KP4    = (KDIM + 3) & ~3;
  constexpr int STRIDE = (((KP4 + 4) & 7) == 0) ? (KP4 + 8) : (KP4 + 4);
  __shared__ float As[16 * STRIDE];

  const int lane = threadIdx.x & 31;
  const int wave = threadIdx.x >> 5;
  const int row0 = blockIdx.x * 16;
  const int m    = lane & 15;
  const int half = lane >> 4;                    // A/B K-pair selector
  const int colb = blockIdx.y * 256 + wave * 32; // 32 output cols per wave
  const int col0 = colb + m;
  const int col1 = colb + 16 + m;

  // Stage the shared 16 x KDIM A-tile into LDS (zero-padded to KP4)
  for (int idx = threadIdx.x; idx < 16 * KP4; idx += 256) {
    int r = idx / KP4;
    int k = idx - r * KP4;
    float v = 0.0f;
    if (KDIM % 4 == 0 || k < KDIM) v = h[(row0 + r) * KDIM + k];
    As[r * STRIDE + k] = v;
  }
  __syncthreads();

  const float* wp0 = Wn + col0;
  const float* wp1 = Wn + col1;
  v8f c0 = {}, c1 = {};
#pragma unroll
  for (int k0 = 0; k0 < KP4; k0 += 4) {
    const int kk = k0 + 2 * half;
    v2f a, b0, b1;
    a.x  = As[m * STRIDE + kk + 0];
    a.y  = As[m * STRIDE + kk + 1];
    b0.x = (KDIM % 4 == 0 || kk     < KDIM) ? wp0[(kk    ) * HD] : 0.0f;
    b0.y = (KDIM % 4 == 0 || kk + 1 < KDIM) ? wp0[(kk + 1) * HD] : 0.0f;
    b1.x = (KDIM % 4 == 0 || kk     < KDIM) ? wp1[(kk    ) * HD] : 0.0f;
    b1.y = (KDIM % 4 == 0 || kk + 1 < KDIM) ? wp1[(kk + 1) * HD] : 0.0f;
    c0 = __builtin_amdgcn_wmma_f32_16x16x4_f32(false, a, false, b0, (short)0, c0,
                                               false, false);
    c1 = __builtin_amdgcn_wmma_f32_16x16x4_f32(false, a, false, b1, (short)0, c1,
                                               false, false);
  }

  // C/D layout: VGPR r -> M = r (lanes 0-15) / M = r+8 (lanes 16-31), N = lane&15
#pragma unroll
  for (int r = 0; r < 8; ++r) {
    const int rr = row0 + r + 8 * half;
    ht[rr * HD + col0] = c0[r];
    ht[rr * HD + col1] = c1[r];
  }
}

// ---------------------------------------------------------------------------
// s_src[n,h] = sum_d ht[n,h,d]*as[h,d]; s_dst likewise with ad.
// ---------------------------------------------------------------------------
__global__ __launch_bounds__(256)
void node_scores(const float* __restrict__ ht, const float* __restrict__ as_,
                 const float* __restrict__ ad_, float* __restrict__ ssrc,
                 float* __restrict__ sdst)
{
  int i = blockIdx.x * blockDim.x + threadIdx.x;
  if (i >= NN * HHC) return;
  int node = i >> 3, hh = i & 7;
  const float4* p  = reinterpret_cast<const float4*>(ht  + (size_t)node * HD + hh * DDC);
  const float4* a1 = reinterpret_cast<const float4*>(as_ + hh * DDC);
  const float4* a2 = reinterpret_cast<const float4*>(ad_ + hh * DDC);
  float acc1 = 0.0f, acc2 = 0.0f;
#pragma unroll 4
  for (int d = 0; d < DDC / 4; ++d) {
    float4 v = p[d], w = a1[d], x = a2[d];
    acc1 += v.x * w.x + v.y * w.y + v.z * w.z + v.w * w.w;
    acc2 += v.x * x.x + v.y * x.y + v.z * x.z + v.w * x.w;
  }
  ssrc[i] = acc1;
  sdst[i] = acc2;
}

// ---------------------------------------------------------------------------
// attn[e,h] = lrelu(s_src[src]+s_dst[dst]+e*We[h]); segment max into mmax[dst,h]
// ---------------------------------------------------------------------------
__global__ __launch_bounds__(256)
void edge_attn_max(const float* __restrict__ ssrc, const float* __restrict__ sdst,
                   const float* __restrict__ efeat, const float* __restrict__ We,
                   const int* __restrict__ src, const int* __restrict__ dst,
                   float* __restrict__ attn, unsigned* __restrict__ mmax)
{
  int i = blockIdx.x * blockDim.x + threadIdx.x;
  if (i >= EE * HHC) return;
  int e = i >> 3, hh = i & 7;
  int s = src[e], t = dst[e];
  float a = ssrc[s * HHC + hh] + sdst[t * HHC + hh] + efeat[e] * We[hh];
  a = lrelu_f(a);
  attn[i] = a;
  atomicMax(&mmax[t * HHC + hh], ford(a));
}

// ---------------------------------------------------------------------------
// ex = exp(attn - m[dst]);  denom[dst,h] += ex  (ex overwrites attn)
// ---------------------------------------------------------------------------
__global__ __launch_bounds__(256)
void edge_exp(float* __restrict__ attn, const unsigned* __restrict__ mmax,
              float* __restrict__ denom, const int* __restrict__ dst)
{
  int i = blockIdx.x * blockDim.x + threadIdx.x;
  if (i >= EE * HHC) return;
  int e = i >> 3, hh = i & 7;
  int t = dst[e];
  float ex = __expf(attn[i] - fdec(mmax[t * HHC + hh]));
  attn[i] = ex;
  atomicAdd(&denom[t * HHC + hh], ex);
}

// ---------------------------------------------------------------------------
// per-edge message, fused head-sum:  hnext[dst,d] += sum_h alpha[h]*ht[src,h,d]
// one 128-thread block per edge (ht gathers hit the 192MB L2: ht is 82MB)
// ---------------------------------------------------------------------------
__global__ __launch_bounds__(128)
void edge_scatter(const float* __restrict__ ht, const float* __restrict__ ex,
                  const float* __restrict__ denom, const int* __restrict__ src,
                  const int* __restrict__ dst, float* __restrict__ hnext)
{
  __shared__ float alpha[HHC];
  int e = blockIdx.x;
  int d = threadIdx.x;
  int s = src[e], t = dst[e];
  if (d < HHC) alpha[d] = ex[e * HHC + d] / denom[t * HHC + d];
  __syncthreads();
  const float* p = ht + (size_t)s * HD + d;
  float acc = 0.0f;
#pragma unroll
  for (int hh = 0; hh < HHC; ++hh) acc += alpha[hh] * p[hh * DDC];
  atomicAdd(&hnext[t * DDC + d], acc);
}

// hcur = lrelu(hnext / H)
__global__ __launch_bounds__(256)
void finalize_k(const float* __restrict__ hnext, float* __restrict__ hcur)
{
  int i = blockIdx.x * blockDim.x + threadIdx.x;
  if (i >= NN * DDC) return;
  hcur[i] = lrelu_f(hnext[i] * 0.125f);
}

// graph pooling: gid == repeat(arange(B), NPG) -> contiguous row ranges
__global__ __launch_bounds__(128)
void pool_graph(const float* __restrict__ h, float* __restrict__ pooled)
{
  int g = blockIdx.x, d = threadIdx.x;
  const float* p = h + (size_t)g * NPGC * DDC + d;
  float acc = 0.0f;
  for (int r = 0; r < NPGC; ++r) acc += p[r * DDC];
  pooled[g * DDC + d] = acc;
}

// hgcat[b, c*64+j] = pooled[b,:] @ Wg1[:,j] + bg1[j]   (no activation, per ref)
__global__ __launch_bounds__(256)
void pool_head(const float* __restrict__ pooled, const float* __restrict__ Wg1,
               const float* __restrict__ bg1, float* __restrict__ hgcat, int c)
{
  int i = blockIdx.x * blockDim.x + threadIdx.x;
  if (i >= BB * 64) return;
  int b = i / 64, j = i % 64;
  float acc = bg1[j];
  for (int k = 0; k < DDC; ++k) acc += pooled[b * DDC + k] * Wg1[k * 64 + j];
  hgcat[b * 406 + c * 64 + j] = acc;
}

// hr = lrelu(ratio@Wr+br) -> cols 384..389 ; htp = lrelu(temp@Wt+bt) -> 390..405
__global__ __launch_bounds__(256)
void small_feats(const float* __restrict__ ratio, const float* __restrict__ Wr,
                 const float* __restrict__ br, const float* __restrict__ temp,
                 const float* __restrict__ Wt, const float* __restrict__ bt,
                 float* __restrict__ hgcat)
{
  int i = blockIdx.x * blockDim.x + threadIdx.x;
  if (i < BB * 6) {
    int b = i / 6, j = i % 6;
    float acc = br[j];
    for (int k = 0; k < 6; ++k) acc += ratio[b * 6 + k] * Wr[k * 6 + j];
    hgcat[b * 406 + 384 + j] = lrelu_f(acc);
  } else if (i < BB * 6 + BB * 16) {
    int r = i - BB * 6;
    int b = r / 16, j = r % 16;
    float acc = bt[j] + temp[b] * Wt[j];
    hgcat[b * 406 + 390 + j] = lrelu_f(acc);
  }
}

__global__ __launch_bounds__(256)
void mlp1(const float* __restrict__ hg, const float* __restrict__ W1,
          const float* __restrict__ b1, float* __restrict__ hidden)
{
  int i = blockIdx.x * blockDim.x + threadIdx.x;
  if (i >= BB * 512) return;
  int b = i / 512, j = i % 512;
  float acc = b1[j];
  for (int k = 0; k < 406; ++k) acc += hg[b * 406 + k] * W1[k * 512 + j];
  hidden[i] = lrelu_f(acc);
}

__global__ __launch_bounds__(128)
void mlp2(const float* __restrict__ hidden, const float* __restrict__ W2,
          const float* __restrict__ b2, float* __restrict__ out)
{
  int b = threadIdx.x;
  if (b >= BB) return;
  float acc = b2[0];
  for (int k = 0; k < 512; ++k) acc += hidden[b * 512 + k] * W2[k];
  out[b] = acc;
}

// ---------------------------------------------------------------------------
extern "C" void kernel_launch(void* const* d_in, const int* in_sizes, int n_in,
                              void* d_out, int out_size, void* d_ws, size_t ws_size,
                              hipStream_t stream)
{
  (void)in_sizes; (void)n_in; (void)out_size; (void)ws_size;

  // setup_inputs() dict order: h1,e1,...,h6,e6, src,dst,gid, ratio,temp,
  // then params leaves: 6x{Wn,We,as,ad}, Wg1,bg1,Wr,br,Wt,bt,W1,b1,W2,b2
  const int*   src   = (const int*)d_in[12];
  const int*   dst   = (const int*)d_in[13];
  const float* ratio = (const float*)d_in[15];
  const float* temp  = (const float*)d_in[16];
  const float* Wg1   = (const float*)d_in[41];
  const float* bg1   = (const float*)d_in[42];
  const float* Wr    = (const float*)d_in[43];
  const float* br    = (const float*)d_in[44];
  const float* Wt    = (const float*)d_in[45];
  const float* bt    = (const float*)d_in[46];
  const float* W1    = (const float*)d_in[47];
  const float* b1    = (const float*)d_in[48];
  const float* W2    = (const float*)d_in[49];
  const float* b2    = (const float*)d_in[50];

  // bump allocator over workspace (~109 MB total)
  char* ws = (char*)d_ws;
  size_t off = 0;
  auto take = [&](size_t bytes) -> char* {
    char* p = ws + off;
    off += (bytes + 255) & ~(size_t)255;
    return p;
  };
  float*    ht     = (float*)take((size_t)NN * HD  * 4);   // 81.9 MB
  float*    hcur   = (float*)take((size_t)NN * DDC * 4);   // 10.2 MB
  float*    hnext  = (float*)take((size_t)NN * DDC * 4);   // 10.2 MB
  float*    ssrc   = (float*)take((size_t)NN * HHC * 4);
  float*    sdst   = (float*)take((size_t)NN * HHC * 4);
  unsigned* mmax   = (unsigned*)take((size_t)NN * HHC * 4);
  float*    denom  = (float*)take((size_t)NN * HHC * 4);
  float*    attn   = (float*)take((size_t)EE * HHC * 4);   // 3.84 MB
  float*    pooled = (float*)take((size_t)BB * DDC * 4);
  float*    hgcat  = (float*)take((size_t)BB * 406 * 4);
  float*    hidden = (float*)take((size_t)BB * 512 * 4);

  for (int c = 0; c < 6; ++c) {
    const float* efeat = (const float*)d_in[2 * c + 1];
    for (int l = 0; l < 6; ++l) {
      const float* Wn  = (const float*)d_in[17 + 4 * l];
      const float* We  = (const float*)d_in[18 + 4 * l];
      const float* as_ = (const float*)d_in[19 + 4 * l];
      const float* ad_ = (const float*)d_in[20 + 4 * l];

      if (l == 0) {
        gat_node_gemm<49><<<dim3(NN / 16, 4), 256, 0, stream>>>(
            (const float*)d_in[2 * c], Wn, ht);
      } else {
        gat_node_gemm<128><<<dim3(NN / 16, 4), 256, 0, stream>>>(hcur, Wn, ht);
      }
      node_scores<<<(NN * HHC + 255) / 256, 256, 0, stream>>>(ht, as_, ad_, ssrc, sdst);

      hipMemsetAsync(mmax, 0, (size_t)NN * HHC * 4, stream);
      hipMemsetAsync(denom, 0, (size_t)NN * HHC * 4, stream);
      edge_attn_max<<<(EE * HHC + 255) / 256, 256, 0, stream>>>(ssrc, sdst, efeat, We,
                                                                src, dst, attn, mmax);
      edge_exp<<<(EE * HHC + 255) / 256, 256, 0, stream>>>(attn, mmax, denom, dst);

      hipMemsetAsync(hnext, 0, (size_t)NN * DDC * 4, stream);
      edge_scatter<<<EE, 128, 0, stream>>>(ht, attn, denom, src, dst, hnext);
      finalize_k<<<(NN * DDC + 255) / 256, 256, 0, stream>>>(hnext, hcur);
    }
    pool_graph<<<BB, 128, 0, stream>>>(hcur, pooled);
    pool_head<<<(BB * 64 + 255) / 256, 256, 0, stream>>>(pooled, Wg1, bg1, hgcat, c);
  }

  small_feats<<<(BB * 6 + BB * 16 + 255) / 256, 256, 0, stream>>>(ratio, Wr, br,
                                                                  temp, Wt, bt, hgcat);
  mlp1<<<(BB * 512 + 255) / 256, 256, 0, stream>>>(hgcat, W1, b1, hidden);
  mlp2<<<1, 128, 0, stream>>>(hidden, W2, b2, (float*)d_out);
}